// remove_padding_75806172774776
// MI455X (gfx1250) — compile-verified
//
#include <hip/hip_runtime.h>

// Problem constants from the reference: x is [B, T, D] float32.
#define B_ 64
#define T_ 4096
#define D_ 128

typedef float v4f __attribute__((ext_vector_type(4)));

#define AS_GLOBAL __attribute__((address_space(1)))
#define AS_LDS    __attribute__((address_space(3)))

// gfx1250 async global->LDS path (ASYNCcnt-tracked); guarded since builtin
// naming/arity varies across toolchains.  Probe round 2 confirmed it exists
// here and takes typed (int*) pointers: (global int*, lds int*, imm, imm).
#if __has_builtin(__builtin_amdgcn_global_load_async_to_lds_b32)
#define HAVE_ASYNC_LDS 1
#else
#define HAVE_ASYNC_LDS 0
#endif

__device__ __forceinline__ void wait_asynccnt0() {
#if __has_builtin(__builtin_amdgcn_s_wait_asynccnt)
  __builtin_amdgcn_s_wait_asynccnt(0);
#else
  asm volatile("s_wait_asynccnt 0" ::: "memory");
#endif
}

// ---------------------------------------------------------------------------
// Kernel 1: per batch row, begin = (first i with x[b,i,0] != x[b,i+1,0],
// else T-1) + 1.  One 256-thread block (8 wave32) per row.
// Stage the channel-0 column (4096 floats = 16KB) into LDS once via
// GLOBAL_LOAD_ASYNC_TO_LDS_B32 (no VGPR round trip), fence on ASYNCcnt +
// workgroup barrier, then do the neighbor compare out of LDS (bank-conflict-
// free: i = lane + 256k and i+1 are both stride-1 across the 64 banks).
// ---------------------------------------------------------------------------
__global__ __launch_bounds__(256) void find_begin_kernel(
    const float* __restrict__ x,
    int* __restrict__ begin_ws,
    float* __restrict__ lengths_out) {
  const int b = blockIdx.x;
  const int tid = threadIdx.x;
  const float* __restrict__ row = x + (size_t)b * T_ * D_;

  __shared__ float s_ch0[T_];
  __shared__ int smin;
  if (tid == 0) smin = 0x7FFFFFFF;

#if HAVE_ASYNC_LDS
#pragma unroll
  for (int k = 0; k < T_ / 256; ++k) {
    const int i = tid + k * 256;
    __builtin_amdgcn_global_load_async_to_lds_b32(
        (AS_GLOBAL int*)(row + (size_t)i * D_),
        (AS_LDS int*)(s_ch0 + i),
        /*offset=*/0, /*cpol=*/0);
  }
  wait_asynccnt0();   // this wave's async LDS writes complete
  __syncthreads();    // all waves' writes visible block-wide
#else
#pragma unroll
  for (int k = 0; k < T_ / 256; ++k) {
    const int i = tid + k * 256;
    s_ch0[i] = row[(size_t)i * D_];
  }
  __syncthreads();
#endif

  int local_min = 0x7FFFFFFF;
#pragma unroll
  for (int k = 0; k < T_ / 256; ++k) {
    const int i = tid + k * 256;
    if (i < T_ - 1 && s_ch0[i] != s_ch0[i + 1] && i < local_min) local_min = i;
  }
  if (local_min != 0x7FFFFFFF) atomicMin(&smin, local_min);
  __syncthreads();

  if (tid == 0) {
    const int m = smin;
    const int first = (m < T_ - 1) ? m : (T_ - 1);  // where(any_neq, first_neq, T-1)
    const int begin = first + 1;
    begin_ws[b] = begin;
    lengths_out[b] = (float)(T_ - begin);
  }
}

// ---------------------------------------------------------------------------
// Kernel 2: shifted streaming copy.  out[b, j, :] = x[b, begin+j, :] when
// begin+j < T, else 0.  One float4 (16B) per lane -> one wave32 moves exactly
// one 512B D-row; begin*D*4 bytes is a multiple of 512B so shifted loads stay
// 16B/cacheline aligned.  Non-temporal TH on both ~134MB streams (read-once /
// write-once) + global_prefetch_b8 read-ahead.  This kernel sets the runtime:
// ~268MB @ 23.3 TB/s ~= 11.5us floor; no FLOPs -> WMMA inapplicable.
// ---------------------------------------------------------------------------
__global__ __launch_bounds__(256) void shift_copy_kernel(
    const float* __restrict__ x,
    const int* __restrict__ begin_ws,
    float* __restrict__ out) {
  constexpr int Q_PER_ROW = T_ * D_ / 4;          // float4 per batch row = 131072
  constexpr int BLOCKS_PER_B = Q_PER_ROW / 256;   // 512
  constexpr int QD = D_ / 4;                      // float4 per D-row = 32

  const int b = blockIdx.x / BLOCKS_PER_B;        // uniform per block -> SGPR
  const int inner = (blockIdx.x % BLOCKS_PER_B) * 256 + threadIdx.x;
  const int j = inner >> 5;                       // time index within output row
  const int q = inner & (QD - 1);                 // float4 index within D

  const int begin = begin_ws[b];                  // scalar (s_load) per block
  const int src = begin + j;

  const v4f* __restrict__ xv = (const v4f*)x;
  v4f* __restrict__ ov = (v4f*)out;

  const size_t out_idx = (size_t)b * Q_PER_ROW + (size_t)inner;
  v4f v = {0.0f, 0.0f, 0.0f, 0.0f};
  if (src < T_) {
    const size_t in_idx = ((size_t)b * T_ + (size_t)src) * QD + (size_t)q;
    v = __builtin_nontemporal_load(&xv[in_idx]);
    // Speculative read-ahead ~32KB into the same stream.
    const size_t pf = in_idx + 2048;
    if (pf < (size_t)B_ * Q_PER_ROW) {
      __builtin_prefetch((const void*)&xv[pf], 0, 0);  // -> global_prefetch_b8
    }
  }
  __builtin_nontemporal_store(v, &ov[out_idx]);
}

// ---------------------------------------------------------------------------
// Host-side launch.  d_in[0]: x [B*T*D] fp32.  d_out: out [B*T*D] fp32 then
// lengths [B] (written as float).  d_ws: >= B ints of scratch for begin[].
// ---------------------------------------------------------------------------
extern "C" void kernel_launch(void* const* d_in, const int* in_sizes, int n_in,
                              void* d_out, int out_size, void* d_ws, size_t ws_size,
                              hipStream_t stream) {
  (void)in_sizes; (void)n_in; (void)out_size; (void)ws_size;

  const float* x = (const float*)d_in[0];
  float* out = (float*)d_out;
  float* lengths = out + (size_t)B_ * T_ * D_;
  int* begin_ws = (int*)d_ws;

  find_begin_kernel<<<B_, 256, 0, stream>>>(x, begin_ws, lengths);

  constexpr int BLOCKS_PER_B = (T_ * D_ / 4) / 256;  // 512
  shift_copy_kernel<<<B_ * BLOCKS_PER_B, 256, 0, stream>>>(x, begin_ws, out);
}